// WordnetFineTuning_50835232916095
// MI455X (gfx1250) — compile-verified
//
#include <hip/hip_runtime.h>
#include <math.h>

// Problem constants (match reference)
#define DIM   384
#define BATCH 4096
#define NEGS  16
#define GNUM  (BATCH * NEGS)

typedef __attribute__((ext_vector_type(2))) float v2f;
typedef __attribute__((ext_vector_type(8))) float v8f;

// ---------------------------------------------------------------------------
// K1/K2: mark segment starts from a sorted segment-id array.
// start[nseg] = n sentinel so count(b) = start[b+1]-start[b].
// (All segments are present with size 4 or 12 by construction.)
// ---------------------------------------------------------------------------
__global__ void mark_starts_kernel(const int* __restrict__ seg, int n, int nseg,
                                   int* __restrict__ start) {
    int i = blockIdx.x * blockDim.x + threadIdx.x;
    if (i >= n) return;
    int s = seg[i];
    if (i == 0 || seg[i - 1] != s) start[s] = i;
    if (i == n - 1) start[nseg] = n;
}

// ---------------------------------------------------------------------------
// K3: one wave per synset b. Pass 1: sum member rows -> centroid (stored).
// Pass 2 (L2 hits): 0.5 * sum ||row - centroid||^2 -> pos_loss[b].
// Lane d-parallel: lane owns dims {lane*4 + 128*j .. +3}, j=0..2 (3x float4).
// ---------------------------------------------------------------------------
__global__ __launch_bounds__(256) void pos_kernel(
    const float* __restrict__ emb, const int* __restrict__ ids,
    const int* __restrict__ start, float* __restrict__ cent,
    float* __restrict__ pos_loss) {
    const int wave = threadIdx.x >> 5;
    const int lane = threadIdx.x & 31;
    const int b = blockIdx.x * 8 + wave;
    const int s = start[b], e = start[b + 1];
    const int col = lane * 4;

    float4 a0 = make_float4(0.f, 0.f, 0.f, 0.f), a1 = a0, a2 = a0;
    for (int i = s; i < e; ++i) {
        const float* row = emb + (size_t)ids[i] * DIM;
        if (i + 1 < e) __builtin_prefetch(emb + (size_t)ids[i + 1] * DIM + col, 0, 3);
        a0 += *(const float4*)(row + col);
        a1 += *(const float4*)(row + col + 128);
        a2 += *(const float4*)(row + col + 256);
    }
    const float invc = 1.0f / (float)(e - s);
    float4 c0 = a0 * invc, c1 = a1 * invc, c2 = a2 * invc;
    float* cb = cent + (size_t)b * DIM + col;
    *(float4*)(cb)       = c0;
    *(float4*)(cb + 128) = c1;
    *(float4*)(cb + 256) = c2;

    float p = 0.f;
    for (int i = s; i < e; ++i) {
        const float* row = emb + (size_t)ids[i] * DIM;
        float4 d0 = *(const float4*)(row + col)       - c0;
        float4 d1 = *(const float4*)(row + col + 128) - c1;
        float4 d2 = *(const float4*)(row + col + 256) - c2;
        p += d0.x * d0.x + d0.y * d0.y + d0.z * d0.z + d0.w * d0.w;
        p += d1.x * d1.x + d1.y * d1.y + d1.z * d1.z + d1.w * d1.w;
        p += d2.x * d2.x + d2.y * d2.y + d2.z * d2.z + d2.w * d2.w;
    }
    for (int off = 16; off > 0; off >>= 1) p += __shfl_xor(p, off, 32);
    if (lane == 0) pos_loss[b] = 0.5f * p;
}

// ---------------------------------------------------------------------------
// K4: one block (16 waves) per synset b. Wave w builds the centroid of
// negative group g = b*16+w, forms T[w][:] = neg_cent - syn_cent in LDS
// (row stride 388 dwords -> conflict-free WMMA feed). Wave 0 then computes
// the 16x16 Gram matrix T*T^T with V_WMMA_F32_16X16X4_F32 (96 k-steps);
// its diagonal is the 16 squared centroid distances -> hinge loss.
// A layout (16x4 f32): lane l, comp j = T[l&15][kb + 2*(l>>4) + j].
// B = T^T (4x16 f32) has the *same* per-lane contents by symmetry.
// ---------------------------------------------------------------------------
__global__ __launch_bounds__(512) void neg_kernel(
    const float* __restrict__ emb, const int* __restrict__ ids,
    const int* __restrict__ nstart, const float* __restrict__ cent,
    const int* __restrict__ margins, float* __restrict__ neg_loss) {
    __shared__ float T[16][388];  // 388-dword row stride: (4*m + col) banks, no conflicts

    const int b    = blockIdx.x;
    const int wave = threadIdx.x >> 5;
    const int lane = threadIdx.x & 31;
    const int g    = (b << 4) + wave;
    const int s = nstart[g], e = nstart[g + 1];
    const int col = lane * 4;

    float4 a0 = make_float4(0.f, 0.f, 0.f, 0.f), a1 = a0, a2 = a0;
    for (int i = s; i < e; ++i) {
        const float* row = emb + (size_t)ids[i] * DIM;
        if (i + 1 < e) __builtin_prefetch(emb + (size_t)ids[i + 1] * DIM + col, 0, 3);
        a0 += *(const float4*)(row + col);
        a1 += *(const float4*)(row + col + 128);
        a2 += *(const float4*)(row + col + 256);
    }
    const float invc = 1.0f / (float)(e - s);
    const float* sc = cent + (size_t)b * DIM + col;
    float4 t0 = a0 * invc - *(const float4*)(sc);
    float4 t1 = a1 * invc - *(const float4*)(sc + 128);
    float4 t2 = a2 * invc - *(const float4*)(sc + 256);
    *(float4*)&T[wave][col]       = t0;
    *(float4*)&T[wave][col + 128] = t1;
    *(float4*)&T[wave][col + 256] = t2;
    __syncthreads();

    if (wave == 0) {  // wave-uniform branch: EXEC all ones inside (WMMA requirement)
        const int m = lane & 15;
        const int h = lane >> 4;
        v8f c = {};
#pragma unroll 4
        for (int kb = 0; kb < DIM; kb += 4) {
            const int k = kb + 2 * h;
            v2f a;
            a.x = T[m][k];
            a.y = T[m][k + 1];
            // D = A(16x4) * B(4x16) + C ; A and B share per-lane contents (Gram)
            c = __builtin_amdgcn_wmma_f32_16x16x4_f32(
                /*neg_a=*/false, a, /*neg_b=*/false, a,
                /*c_mod=*/(short)0, c, /*reuse_a=*/false, /*reuse_b=*/false);
        }
        // Diagonal of C: M<8 -> VGPR M lane M ; M>=8 -> VGPR M-8 lane 16+M
        float d2[NEGS];
#pragma unroll
        for (int n = 0; n < NEGS; ++n) {
            float src = (n < 8) ? c[n] : c[n - 8];
            d2[n] = __shfl(src, (n < 8) ? n : (16 + n), 32);
        }
        if (lane == 0) {
            const int* mg = margins + b * NEGS;
            float tot = 0.f;
#pragma unroll
            for (int n = 0; n < NEGS; ++n) {
                float dist = sqrtf(fmaxf(d2[n], 0.f));
                float hh = fmaxf((float)mg[n] - dist, 0.f);
                tot += 0.5f * hh * hh;
            }
            neg_loss[b] = tot;
        }
    }
}

// ---------------------------------------------------------------------------
// K5: deterministic single-block reduction -> mean over BATCH.
// ---------------------------------------------------------------------------
__global__ __launch_bounds__(256) void final_reduce_kernel(
    const float* __restrict__ pos_loss, const float* __restrict__ neg_loss,
    float* __restrict__ out) {
    __shared__ float sm[256];
    float p = 0.f;
    for (int i = threadIdx.x; i < BATCH; i += 256) p += pos_loss[i] + neg_loss[i];
    sm[threadIdx.x] = p;
    __syncthreads();
    for (int off = 128; off > 0; off >>= 1) {
        if ((int)threadIdx.x < off) sm[threadIdx.x] += sm[threadIdx.x + off];
        __syncthreads();
    }
    if (threadIdx.x == 0) out[0] = sm[0] / (float)BATCH;
}

// ---------------------------------------------------------------------------
extern "C" void kernel_launch(void* const* d_in, const int* in_sizes, int n_in,
                              void* d_out, int out_size, void* d_ws, size_t ws_size,
                              hipStream_t stream) {
    const float* emb     = (const float*)d_in[0];
    const int* pos_ids   = (const int*)d_in[1];
    const int* pos_seg   = (const int*)d_in[2];
    const int* neg_ids   = (const int*)d_in[3];
    const int* neg_seg   = (const int*)d_in[4];
    const int* margins   = (const int*)d_in[5];
    const int NP = in_sizes[1];
    const int NN = in_sizes[3];

    // workspace carve-out (256B aligned); every byte read is written each call
    char* w = (char*)d_ws;
    auto carve = [&](size_t bytes) {
        char* p = w;
        w += (bytes + 255) & ~(size_t)255;
        return p;
    };
    int*   pos_start = (int*)carve((size_t)(BATCH + 1) * sizeof(int));
    int*   neg_start = (int*)carve((size_t)(GNUM + 1) * sizeof(int));
    float* cent      = (float*)carve((size_t)BATCH * DIM * sizeof(float));
    float* pos_loss  = (float*)carve((size_t)BATCH * sizeof(float));
    float* neg_loss  = (float*)carve((size_t)BATCH * sizeof(float));

    mark_starts_kernel<<<(NP + 255) / 256, 256, 0, stream>>>(pos_seg, NP, BATCH, pos_start);
    mark_starts_kernel<<<(NN + 255) / 256, 256, 0, stream>>>(neg_seg, NN, GNUM, neg_start);
    pos_kernel<<<BATCH / 8, 256, 0, stream>>>(emb, pos_ids, pos_start, cent, pos_loss);
    neg_kernel<<<BATCH, 512, 0, stream>>>(emb, neg_ids, neg_start, cent, margins, neg_loss);
    final_reduce_kernel<<<1, 256, 0, stream>>>(pos_loss, neg_loss, (float*)d_out);
}